// SO2Convolution_49435073577112
// MI455X (gfx1250) — compile-verified
//
#include <hip/hip_runtime.h>

typedef float v2f __attribute__((ext_vector_type(2)));
typedef float v4f __attribute__((ext_vector_type(4)));
typedef float v8f __attribute__((ext_vector_type(8)));

#define WAVES 12
#define TPB (WAVES * 32)

constexpr int NB   = 20000;  // edges
constexpr int ECH  = 128;    // edge channels
constexpr int CIN  = 64;
constexpr int TOUT = 320;    // out+gate channels
constexpr int ROUT = 1920;   // radial MLP output
constexpr int XSTR = 1920;   // per-edge floats in x / rad  (5*3*2*64)
constexpr int OSTR = 9600;   // per-edge floats in out      (5*3*2*320)
constexpr int SLAB = 640;    // per-(l,m) floats in out     (2*320)

// LDS layout (floats): xe[16*128] | h[16*128] | scr[288] | wbuf[12][16*128]
constexpr int SMEM_FLOATS = 16 * ECH + 16 * ECH + 288 + WAVES * 16 * ECH;

__global__ __launch_bounds__(TPB) void so2_fused(
    const float* __restrict__ x,   const float* __restrict__ xe_g,
    const float* __restrict__ Wr,  const float* __restrict__ Wi,
    const float* __restrict__ w1,  const float* __restrict__ b1,
    const float* __restrict__ lng, const float* __restrict__ lnb,
    const float* __restrict__ w2,  const float* __restrict__ b2,
    float* __restrict__ out)
{
    extern __shared__ float smem[];
    float* xe   = smem;                 // 16 x 128 edge-feature tile
    float* h    = xe + 16 * ECH;        // 16 x 128 hidden
    float* scr  = h + 16 * ECH;         // 256 partials + 16 mu + 16 var
    float* wbuf = scr + 288;            // per-wave 16 x 128 rad/xs buffer

    const int tid  = threadIdx.x;
    const int w    = tid >> 5;          // wave id (wave32)
    const int lane = tid & 31;
    const int half = lane >> 4;         // K-half selector for fp32 WMMA frags
    const int t16  = lane & 15;
    const int b0   = blockIdx.x * 16;   // 20000 % 16 == 0 -> no guards

    // ---- stage x_edge tile into LDS ----------------------------------------
    for (int i = tid; i < 16 * ECH; i += TPB)
        xe[i] = xe_g[(size_t)(b0 + (i >> 7)) * ECH + (i & 127)];
    __syncthreads();

    // ---- GEMM1: h = xe @ w1 + b1  (waves 0..7, one 16x16 N-tile each) ------
    if (w < 8) {
        const int n0 = w * 16;
        const float bv = b1[n0 + t16];
        v8f c;
        for (int v = 0; v < 8; ++v) c[v] = bv;
        for (int k = 0; k < 32; ++k) {
            const int kk = 4 * k + 2 * half;
            v2f a; a.x = xe[t16 * ECH + kk];        a.y = xe[t16 * ECH + kk + 1];
            v2f b; b.x = w1[kk * ECH + n0 + t16];   b.y = w1[(kk + 1) * ECH + n0 + t16];
            c = __builtin_amdgcn_wmma_f32_16x16x4_f32(false, a, false, b,
                                                      (short)0, c, false, false);
        }
        for (int v = 0; v < 8; ++v)
            h[(v + 8 * half) * ECH + n0 + t16] = c[v];
    }
    __syncthreads();

    // ---- LayerNorm + SiLU on h (threads 0..255: 16 threads per edge) -------
    if (tid < 256) {
        const int e = tid >> 4, t = tid & 15;
        float s = 0.f;
        for (int jj = 0; jj < 8; ++jj) s += h[e * ECH + t + 16 * jj];
        scr[tid] = s;
    }
    __syncthreads();
    if (tid < 16) {
        float s = 0.f;
        for (int jj = 0; jj < 16; ++jj) s += scr[tid * 16 + jj];
        scr[256 + tid] = s * (1.0f / ECH);          // mu
    }
    __syncthreads();
    if (tid < 256) {
        const int e = tid >> 4, t = tid & 15;
        const float mu = scr[256 + e];
        float s = 0.f;
        for (int jj = 0; jj < 8; ++jj) {
            const float d = h[e * ECH + t + 16 * jj] - mu;
            s += d * d;
        }
        scr[tid] = s;
    }
    __syncthreads();
    if (tid < 16) {
        float s = 0.f;
        for (int jj = 0; jj < 16; ++jj) s += scr[tid * 16 + jj];
        scr[272 + tid] = s * (1.0f / ECH);          // var
    }
    __syncthreads();
    if (tid < 256) {
        const int e = tid >> 4, t = tid & 15;
        const float mu   = scr[256 + e];
        const float rstd = rsqrtf(scr[272 + e] + 1e-5f);
        for (int jj = 0; jj < 8; ++jj) {
            const int ch = t + 16 * jj;
            float y = (h[e * ECH + ch] - mu) * rstd * lng[ch] + lnb[ch];
            h[e * ECH + ch] = y / (1.0f + __expf(-y));   // SiLU
        }
    }
    __syncthreads();

    // ---- per-wave (l,m) job ------------------------------------------------
    // valid jobs: all j = 3l+m in 0..14 except {1,2,5} (m > l)
    int j = w;
    if (w >= 1) j += 2;   // skip 1,2
    if (w >= 3) j += 1;   // skip 5
    const int l = j / 3, m = j - 3 * l;
    float* wb = wbuf + w * (16 * ECH);

    // fully-masked slabs (m > l): jobs 1,2,5 -> zero-filled by the three
    // lightest waves (1, 3, 6: their own jobs have only 4 einsum tiles)
    if (w == 1 || w == 3 || w == 6) {
        const int zj = (w == 1) ? 1 : ((w == 3) ? 2 : 5);
        const v4f z = {0.f, 0.f, 0.f, 0.f};
        for (int e = 0; e < 16; ++e) {
            v4f* p = (v4f*)(out + (size_t)(b0 + e) * OSTR + zj * SLAB);
            for (int i = lane; i < SLAB / 4; i += 32) p[i] = z;
        }
    }

    // zero-fill only the masked columns of own slab; computed ranges
    // ([0,320) for (0,0), [0,64) for m=0 l>0, [0,64)+[320,384) for m>0)
    // are written exactly once by the einsum stores below.
    {
        const int chi = (m == 0) ? ((l == 0) ? 320 : 64) : 64;  // computed low range
        const v4f z = {0.f, 0.f, 0.f, 0.f};
        for (int e = 0; e < 16; ++e) {
            v4f* p = (v4f*)(out + (size_t)(b0 + e) * OSTR + j * SLAB);
            for (int i = lane; i < SLAB / 4; i += 32) {
                const int col = i << 2;
                const bool computed = (col < chi) ||
                                      (m != 0 && col >= 320 && col < 384);
                if (!computed) p[i] = z;
            }
        }
    }

    // ---- GEMM2 section: wb = h @ w2[:, j*128 : j*128+128] + b2 -------------
    for (int nt = 0; nt < 8; ++nt) {
        const int ncol = j * 128 + nt * 16 + t16;
        const float bv = b2[ncol];
        v8f c;
        for (int v = 0; v < 8; ++v) c[v] = bv;
        for (int k = 0; k < 32; ++k) {
            const int kk = 4 * k + 2 * half;
            v2f a; a.x = h[t16 * ECH + kk];        a.y = h[t16 * ECH + kk + 1];
            v2f b; b.x = w2[kk * ROUT + ncol];     b.y = w2[(kk + 1) * ROUT + ncol];
            c = __builtin_amdgcn_wmma_f32_16x16x4_f32(false, a, false, b,
                                                      (short)0, c, false, false);
        }
        for (int v = 0; v < 8; ++v)
            wb[(v + 8 * half) * ECH + nt * 16 + t16] = c[v];
    }

    // ---- xs = x * rad (in place; 32 consecutive floats per step: coalesced)
    {
        const float* xp = x + (size_t)b0 * XSTR + j * 128;
        for (int i = lane; i < 16 * ECH; i += 32) {
            const int e = i >> 7, cidx = i & 127;
            wb[i] *= xp[(size_t)e * XSTR + cidx];
        }
    }

    // ---- einsum: out[b, l, m, R, o] = sum_{r,c} xs[r,c] * W[m,R,r,c,o] -----
    // W[m,0,0]=Wr, W[m,0,1]=Wi, W[m,1,0]=-Wi, W[m,1,1]=Wr
    // only unmasked N-tiles: (0,0): 20 (out+gate, R=0); m=0,l>0: 4; m>0: 4+4
    const int ntCount = (m == 0) ? ((l == 0) ? 20 : 4) : 8;
    const float* Wrm = Wr + m * CIN * TOUT;
    const float* Wim = Wi + m * CIN * TOUT;
    for (int it = 0; it < ntCount; ++it) {
        const int nb   = (m != 0 && it >= 4) ? (320 + (it - 4) * 16) : it * 16;
        const int ncol = nb + t16;                  // 0..639 within slab
        const int R    = (ncol >= TOUT) ? 1 : 0;
        const int o    = ncol - R * TOUT;
        v8f c;
        for (int v = 0; v < 8; ++v) c[v] = 0.f;
        for (int r = 0; r < 2; ++r) {
            const float* Wbase = ((R + r) & 1) ? Wim : Wrm;
            const float  sgn   = (R == 1 && r == 0) ? -1.0f : 1.0f;
            for (int k = 0; k < 16; ++k) {
                const int cc = 4 * k + 2 * half;    // col within 64-chunk
                const int kk = r * 64 + cc;         // K index 0..127
                v2f a; a.x = wb[t16 * ECH + kk];       a.y = wb[t16 * ECH + kk + 1];
                v2f b; b.x = sgn * Wbase[cc * TOUT + o];
                       b.y = sgn * Wbase[(cc + 1) * TOUT + o];
                c = __builtin_amdgcn_wmma_f32_16x16x4_f32(false, a, false, b,
                                                          (short)0, c, false, false);
            }
        }
        // store: VGPR v holds rows (edge) v and v+8 across lane halves;
        // lanes 0..15 of each half cover 64 contiguous bytes -> coalesced
        for (int v = 0; v < 8; ++v) {
            const int e = v + 8 * half;
            out[(size_t)(b0 + e) * OSTR + j * SLAB + ncol] = c[v];
        }
    }
}

extern "C" void kernel_launch(void* const* d_in, const int* in_sizes, int n_in,
                              void* d_out, int out_size, void* d_ws, size_t ws_size,
                              hipStream_t stream) {
    const float* x   = (const float*)d_in[0];
    const float* xe  = (const float*)d_in[1];
    const float* Wr  = (const float*)d_in[2];
    const float* Wi  = (const float*)d_in[3];
    const float* w1  = (const float*)d_in[4];
    const float* b1  = (const float*)d_in[5];
    const float* lng = (const float*)d_in[6];
    const float* lnb = (const float*)d_in[7];
    const float* w2  = (const float*)d_in[8];
    const float* b2  = (const float*)d_in[9];

    const size_t smem = (size_t)SMEM_FLOATS * sizeof(float);   // ~113 KB < 320 KB/WGP
    so2_fused<<<NB / 16, TPB, smem, stream>>>(x, xe, Wr, Wi, w1, b1, lng, lnb,
                                              w2, b2, (float*)d_out);
}